// SGCN_71262097375396
// MI455X (gfx1250) — compile-verified
//
#include <hip/hip_runtime.h>

// SGCN fused kernel for MI455X (gfx1250, wave32, fp32 WMMA 16x16x4).
//
// out[n,c,t,w] = sum_v ( sum_ci x[n,ci,t,v]*W[ci,c] + b[c] ) * a[n,t,v,w]
// d_out = [ out (64*64*300*25 floats) | a copy (64*300*25*25 floats) ]
//
// All LDS operand tiles are K-pair interleaved so every WMMA A/B operand is a
// single 8B-aligned ds_load_b64 straight into an even-aligned VGPR pair.

typedef __attribute__((ext_vector_type(2))) float v2f;
typedef __attribute__((ext_vector_type(8))) float v8f;

#define N_    64
#define CIN   64
#define COUT  64
#define T_    300
#define V_    25

#define WAVES 4     // one (n,t) site per wave
#define WST   80    // wlds2 row stride (float2): 2*80 mod 64 == 32 -> lane-halves disjoint banks
#define XST   48    // xs2 / as2 row stride (float2): 2*48 mod 64 == 32
#define YST   28    // ys row stride (float): gcd(28,64)=4 -> conflict-free column reads

__global__ __launch_bounds__(32 * WAVES)
void sgcn_fused_wmma(const float* __restrict__ x,
                     const float* __restrict__ a,
                     const float* __restrict__ W,
                     const float* __restrict__ b,
                     float* __restrict__ out)
{
    // K-pair interleaved W: wlds2[p*WST + d] = { W[2p][d], W[2p+1][d] }   (20 KB)
    __shared__ v2f   wlds2[32 * WST];
    __shared__ float blds[COUT];
    __shared__ v2f   xs2[WAVES][32 * XST];      // {X[2p][v], X[2p+1][v]}      12 KB/wave
    __shared__ float ys [WAVES][COUT * YST];    // Y row-major                  7 KB/wave
    __shared__ v2f   as2[WAVES][14 * XST];      // {A[2p][w], A[2p+1][w]}     5.25 KB/wave

    const int tid  = threadIdx.x;
    const int lane = tid & 31;
    const int wave = tid >> 5;
    const int half = lane >> 4;     // WMMA 32-bit operand lane-half
    const int l16  = lane & 15;

    // ---- stage W (K-pair interleaved) and b, once per block ----
    for (int i = tid; i < 32 * COUT; i += 32 * WAVES) {
        int p = i >> 6, d = i & 63;
        v2f w2;
        w2[0] = W[(2 * p) * COUT + d];
        w2[1] = W[(2 * p + 1) * COUT + d];
        wlds2[p * WST + d] = w2;
    }
    if (tid < COUT) blds[tid] = b[tid];
    __syncthreads();

    const int task = blockIdx.x * WAVES + wave;    // 19200 = 4800 blocks * 4 waves
    const int n = task / T_;
    const int t = task - n * T_;

    v2f*   xw2 = xs2[wave];
    float* yw  = ys[wave];
    v2f*   aw2 = as2[wave];

    // ---- load x[n,:,t,:] as channel pairs; zero-pad v=25..31 ----
    {
        const float* xp = x + ((size_t)n * CIN * T_ + t) * V_;
        for (int c = 0; c < CIN; c += 2) {
            v2f v = {0.f, 0.f};
            if (lane < V_) {
                v[0] = xp[(size_t)c * (T_ * V_) + lane];
                v[1] = xp[(size_t)(c + 1) * (T_ * V_) + lane];
            }
            xw2[(c >> 1) * XST + lane] = v;
        }
    }
    // ---- load a[n,t,:,:] pair-interleaved; zero K-pad rows 25..27 (pairs 12,13) ----
    {
        const float* ap  = a + ((size_t)n * T_ + t) * (V_ * V_);
        float*       awf = (float*)aw2;
        for (int i = lane; i < 2 * XST * 2; i += 32) awf[12 * XST * 2 + i] = 0.0f;
        for (int i = lane; i < V_ * V_; i += 32) {
            int v = i / V_, w = i - v * V_;
            awf[((v >> 1) * XST + w) * 2 + (v & 1)] = ap[i];
        }
    }

    // ================= GEMM 1: Y[d][v] = sum_c W[c][d] * X[c][v] + b[d] =================
    v8f acc[4][2];
#pragma unroll
    for (int mt = 0; mt < 4; ++mt) {
#pragma unroll
        for (int r = 0; r < 8; ++r) {
            float bb = blds[mt * 16 + 8 * half + r];   // D.v[r] row = m0 + r + 8*half
            acc[mt][0][r] = bb;
            acc[mt][1][r] = bb;
        }
    }
#pragma unroll 4
    for (int k0 = 0; k0 < CIN; k0 += 4) {
        const int q = (k0 >> 1) + half;                // K-pair index for this lane-half
        v2f B0 = xw2[q * XST + l16];
        v2f B1 = xw2[q * XST + 16 + l16];
#pragma unroll
        for (int mt = 0; mt < 4; ++mt) {
            v2f Av = wlds2[q * WST + mt * 16 + l16];
            acc[mt][0] = __builtin_amdgcn_wmma_f32_16x16x4_f32(
                false, Av, false, B0, (short)0, acc[mt][0], false, false);
            acc[mt][1] = __builtin_amdgcn_wmma_f32_16x16x4_f32(
                false, Av, false, B1, (short)0, acc[mt][1], false, false);
        }
    }

    // ---- spill Y to LDS (D-layout -> row-major; cols 25..27 = bias, killed by zero A rows) ----
#pragma unroll
    for (int mt = 0; mt < 4; ++mt) {
#pragma unroll
        for (int nt = 0; nt < 2; ++nt) {
            int col = nt * 16 + l16;
            if (col < YST) {
#pragma unroll
                for (int r = 0; r < 8; ++r)
                    yw[(mt * 16 + r + 8 * half) * YST + col] = acc[mt][nt][r];
            }
        }
    }

    // ================= GEMM 2: Out[d][w] = sum_v Y[d][v] * A[v][w] =================
    v8f oacc[4][2];
#pragma unroll
    for (int mt = 0; mt < 4; ++mt)
#pragma unroll
        for (int nt = 0; nt < 2; ++nt) {
            v8f z = {0.f, 0.f, 0.f, 0.f, 0.f, 0.f, 0.f, 0.f};
            oacc[mt][nt] = z;
        }
#pragma unroll
    for (int k0 = 0; k0 < 28; k0 += 4) {
        const int q = (k0 >> 1) + half;
        v2f B0 = aw2[q * XST + l16];
        v2f B1 = aw2[q * XST + 16 + l16];
#pragma unroll
        for (int mt = 0; mt < 4; ++mt) {
            // Y[m][k],Y[m][k+1] are adjacent in the row-major tile: one aligned b64 load
            v2f Av = *(const v2f*)&yw[(mt * 16 + l16) * YST + k0 + 2 * half];
            oacc[mt][0] = __builtin_amdgcn_wmma_f32_16x16x4_f32(
                false, Av, false, B0, (short)0, oacc[mt][0], false, false);
            oacc[mt][1] = __builtin_amdgcn_wmma_f32_16x16x4_f32(
                false, Av, false, B1, (short)0, oacc[mt][1], false, false);
        }
    }

    // ---- store out[n, d, t, w]; 16-lane halves give 64B coalesced runs per row ----
    float* op = out + ((size_t)n * COUT * T_ + t) * V_;
#pragma unroll
    for (int mt = 0; mt < 4; ++mt) {
#pragma unroll
        for (int nt = 0; nt < 2; ++nt) {
            int w = nt * 16 + l16;
            if (w < V_) {
#pragma unroll
                for (int r = 0; r < 8; ++r)
                    op[(size_t)(mt * 16 + r + 8 * half) * (T_ * V_) + w] = oacc[mt][nt][r];
            }
        }
    }
}

// second tuple element: verbatim copy of `a` (pure bandwidth, float4)
__global__ void copy_a_f4(const float4* __restrict__ src, float4* __restrict__ dst, int n4)
{
    int i = blockIdx.x * blockDim.x + threadIdx.x;
    if (i < n4) dst[i] = src[i];
}

extern "C" void kernel_launch(void* const* d_in, const int* in_sizes, int n_in,
                              void* d_out, int out_size, void* d_ws, size_t ws_size,
                              hipStream_t stream)
{
    const float* x = (const float*)d_in[0];
    const float* a = (const float*)d_in[1];
    const float* W = (const float*)d_in[2];
    const float* b = (const float*)d_in[3];

    float* out   = (float*)d_out;
    const size_t out_elems = (size_t)N_ * COUT * T_ * V_;   // 30,720,000
    float* a_out = out + out_elems;

    const int tasks  = N_ * T_;                // 19200
    const int blocks = tasks / WAVES;          // 4800
    sgcn_fused_wmma<<<blocks, 32 * WAVES, 0, stream>>>(x, a, W, b, out);

    const int n4 = (N_ * T_ * V_ * V_) / 4;    // 3,000,000 float4
    copy_a_f4<<<(n4 + 255) / 256, 256, 0, stream>>>((const float4*)a, (float4*)a_out, n4);
}